// SparseConv3d_Atomic_88373247082550
// MI455X (gfx1250) — compile-verified
//
#include <hip/hip_runtime.h>

// Sparse rulebook conv3d: gather -> fp32 WMMA GEMM -> fp32 atomic scatter-add.
// MI455X / gfx1250, wave32. One wave = 16 rules x 64 Cout tile.
// Weights staged to LDS via the Tensor Data Mover (hardware pad -> stride-72 swizzle).

typedef __attribute__((ext_vector_type(2))) float v2f;
typedef __attribute__((ext_vector_type(8))) float v8f;
typedef __attribute__((ext_vector_type(4))) unsigned int u32x4;
typedef __attribute__((ext_vector_type(4))) int i32x4;
typedef __attribute__((ext_vector_type(8))) int i32x8;

#define KOFF 27
#define RPK  50000
#define CIN  64
#define COUT 64
#define TILES (RPK / 16)          // 3125 rule-tiles per offset
#define WAVES_PER_BLOCK 4
#define LDS_STRIDE 72             // 64 data + 8 pad dwords; 2*72 mod 64 = 16 -> halves on disjoint banks

#if __has_builtin(__builtin_amdgcn_tensor_load_to_lds) && __has_builtin(__builtin_amdgcn_s_wait_tensorcnt)
#define HAVE_TDM 1
#else
#define HAVE_TDM 0
#endif

__device__ __forceinline__ void atomic_add_f32(float* p, float v) {
#if __has_builtin(__builtin_amdgcn_global_atomic_fadd_f32)
  __builtin_amdgcn_global_atomic_fadd_f32(p, v);   // global_atomic_add_f32, no return
#else
  atomicAdd(p, v);
#endif
}

__global__ void __launch_bounds__(256) init_out_kernel(float* __restrict__ out,
                                                       const float* __restrict__ bias,
                                                       int n) {
  int i = blockIdx.x * blockDim.x + threadIdx.x;
  if (i < n) out[i] = bias[i & (COUT - 1)];
}

__global__ void __launch_bounds__(128) sparse_conv_wmma_kernel(
    const float* __restrict__ x,        // [N_IN, 64]
    const float* __restrict__ w,        // [27, 64, 64]
    const int*   __restrict__ in_idx,   // [27*R]
    const int*   __restrict__ out_idx,  // [27*R]
    float*       __restrict__ out)      // [N_OUT, 64]
{
  __shared__ float wl[CIN * LDS_STRIDE];

  const int k = blockIdx.y;
  const float* __restrict__ wk = w + (size_t)k * (CIN * COUT);

#if HAVE_TDM
  // ---- TDM: DMA W[k] (64x64 fp32) into LDS with 8-dword pad after every 64 dwords ----
  // D# group0: count=1 | lds_addr | global_addr[56:0] | type=2
  // D# group1: data_size=4B, pad_enable, pad_interval=5 (64 dw), pad_amount=7 (8 dw),
  //            tensor_dim0=64, tensor_dim1=64, tile_dim0=64, tile_dim1=64, dim0_stride=64
  if (threadIdx.x < 32) {                     // one wave issues; TDM ignores EXEC
    const unsigned long long ga = (unsigned long long)(uintptr_t)wk;
    const unsigned int lds_b = (unsigned int)(uintptr_t)wl;   // low 32 bits = LDS byte offset
    u32x4 g0;
    g0[0] = 1u;                                                // count=1, user mode
    g0[1] = lds_b;                                             // lds_addr
    g0[2] = (unsigned int)ga;                                  // global_addr[31:0]
    g0[3] = ((unsigned int)(ga >> 32) & 0x01FFFFFFu) | (2u << 30); // ga[56:32] | type=2
    i32x8 g1;
    g1[0] = (int)((2u << 16) | (1u << 20) | (5u << 22) | (7u << 25)); // dsize=4B|pad|int=64dw|amt=8dw
    g1[1] = (int)(64u << 16);   // tensor_dim0[15:0] in [31:16]
    g1[2] = (int)(64u << 16);   // tensor_dim0[31:16]=0 ; tensor_dim1[15:0]=64 in [31:16]
    g1[3] = (int)(64u << 16);   // tensor_dim1[31:16]=0 ; tile_dim0=64 in [31:16]
    g1[4] = 64;                 // tile_dim1=64 ; tile_dim2=0 (2D)
    g1[5] = 64;                 // tensor_dim0_stride[31:0]=64
    g1[6] = 0;                  // dim0_stride[47:32]=0 ; dim1_stride lo (unused, 2D)
    g1[7] = 0;
    i32x4 gz = {0, 0, 0, 0};    // groups 2/3: dims 3/4 unused (tile_dim3/4 = 0)
#if defined(__clang_major__) && __clang_major__ >= 23
    i32x8 gz8 = {0, 0, 0, 0, 0, 0, 0, 0};
    __builtin_amdgcn_tensor_load_to_lds(g0, g1, gz, gz, gz8, 0);
#else
    __builtin_amdgcn_tensor_load_to_lds(g0, g1, gz, gz, 0);
#endif
    __builtin_amdgcn_s_wait_tensorcnt(0);     // TENSORcnt -> 0 before anyone reads LDS
  }
#else
  __builtin_prefetch(wk, 0, 3);
  for (int i = threadIdx.x; i < CIN * COUT; i += blockDim.x) {
    wl[(i >> 6) * LDS_STRIDE + (i & 63)] = wk[i];
  }
#endif
  __syncthreads();

  const int wave = threadIdx.x >> 5;
  const int lane = threadIdx.x & 31;
  const int m    = lane & 15;          // row (A) / col (B,C) within 16
  const int half = lane >> 4;          // which K-pair / C-row-half

  const int t = blockIdx.x * WAVES_PER_BLOCK + wave;
  if (t >= TILES) return;              // whole-wave uniform; EXEC stays all-1s for WMMA

  const long base = (long)k * RPK + (long)t * 16;
  const int irow = in_idx[base + m];   // lanes m and m+16 load the same index
  const int orow = out_idx[base + m];

  // ---- Gather A in native WMMA layout ----
  // A 16x4 fp32 per step kk: lane(l<16):  {K=4kk+0, K=4kk+1}
  //                          lane(l>=16): {K=4kk+2, K=4kk+3}  -> one float2 load each.
  const v2f* __restrict__ ap = (const v2f*)(x + (long)irow * CIN + 2 * half);
  v2f a[16];
  #pragma unroll
  for (int kk = 0; kk < 16; ++kk) a[kk] = ap[2 * kk];

  // ---- 64 x V_WMMA_F32_16X16X4_F32: 4 N-tiles x 16 K-steps ----
  v8f acc[4] = {};
  #pragma unroll
  for (int kk = 0; kk < 16; ++kk) {
    const int rb = 4 * kk + 2 * half;                     // B rows for this lane-half
    const float* __restrict__ wr0 = &wl[rb * LDS_STRIDE + m];
    const float* __restrict__ wr1 = &wl[(rb + 1) * LDS_STRIDE + m];
    #pragma unroll
    for (int nt = 0; nt < 4; ++nt) {
      v2f b;
      b.x = wr0[nt * 16];                                 // B[rb  ][nt*16+m]
      b.y = wr1[nt * 16];                                 // B[rb+1][nt*16+m]
      acc[nt] = __builtin_amdgcn_wmma_f32_16x16x4_f32(
          /*neg_a=*/false, a[kk], /*neg_b=*/false, b,
          /*c_mod=*/(short)0, acc[nt],
          /*reuse_a=*/false, /*reuse_b=*/false);
    }
  }

  // ---- Scatter-add: C/D layout -> lane l vgpr j holds row (j + 8*half), col m ----
  #pragma unroll
  for (int nt = 0; nt < 4; ++nt) {
    #pragma unroll
    for (int j = 0; j < 8; ++j) {
      const int mr  = j + 8 * half;
      const int orj = __shfl(orow, mr, 32);
      atomic_add_f32(out + (long)orj * COUT + nt * 16 + m, acc[nt][j]);
    }
  }
}

extern "C" void kernel_launch(void* const* d_in, const int* in_sizes, int n_in,
                              void* d_out, int out_size, void* d_ws, size_t ws_size,
                              hipStream_t stream) {
  const float* x       = (const float*)d_in[0];
  const float* w       = (const float*)d_in[1];
  const float* bias    = (const float*)d_in[2];
  const int*   in_idx  = (const int*)d_in[3];
  const int*   out_idx = (const int*)d_in[4];
  float*       out     = (float*)d_out;

  // out = bias (also clears poisoned d_out deterministically every call)
  init_out_kernel<<<(out_size + 255) / 256, 256, 0, stream>>>(out, bias, out_size);

  dim3 grid((TILES + WAVES_PER_BLOCK - 1) / WAVES_PER_BLOCK, KOFF);
  sparse_conv_wmma_kernel<<<grid, WAVES_PER_BLOCK * 32, 0, stream>>>(
      x, w, in_idx, out_idx, out);
}